// ROIPooling2dPytorch_28595892257663
// MI455X (gfx1250) — compile-verified
//
#include <hip/hip_runtime.h>

// ---------------- problem constants (from the reference) ----------------
#define OUT_HW   7
#define SCALE    0.0625f
#define NCH      256
#define FH       38
#define FW       38
#define NUM_ROIS 256

#define CHUNK    8          // channels staged per LDS tile
#define THREADS  256        // 8 wave32 per workgroup
#define TILE_ELEMS (CHUNK * FH * FW)   // worst-case 11552 floats = 46.2 KB

typedef __attribute__((address_space(1))) int g_i32;   // global int
typedef __attribute__((address_space(3))) int l_i32;   // LDS int

// ---- CDNA5 async global->LDS copy (ASYNCcnt path), 4 bytes per lane ----
__device__ __forceinline__ void async_copy_b32(const float* gsrc, float* ldst) {
#if __has_builtin(__builtin_amdgcn_global_load_async_to_lds_b32)
    __builtin_amdgcn_global_load_async_to_lds_b32(
        (g_i32*)gsrc, (l_i32*)ldst, /*imm offset*/0, /*cpol*/0);
#else
    asm volatile("global_load_async_to_lds_b32 %0, %1, off"
                 :
                 : "v"((l_i32*)ldst), "v"(gsrc)
                 : "memory");
#endif
}

__device__ __forceinline__ void wait_async_zero() {
#if __has_builtin(__builtin_amdgcn_s_wait_asynccnt)
    __builtin_amdgcn_s_wait_asynccnt(0);
#else
    asm volatile("s_wait_asynccnt 0" ::: "memory");
#endif
}

// Issue async copies for one CHUNK x rh x rw tile (no wait).
__device__ __forceinline__ void stage_chunk(const float* __restrict__ feat,
                                            int b, int c0, int y1, int x1,
                                            int rh, int rw, int area, int nelem,
                                            float* buf, int tid) {
    const float* gbase = feat + (((size_t)b * NCH + c0) * FH + y1) * FW + x1;
    for (int t = tid; t < nelem; t += THREADS) {
        const int cc  = t / area;
        const int rem = t - cc * area;
        const int yy  = rem / rw;
        const int xx  = rem - yy * rw;
        async_copy_b32(gbase + ((size_t)cc * FH + yy) * FW + xx, &buf[t]);
    }
}

__global__ __launch_bounds__(THREADS)
void roi_max_pool_kernel(const float* __restrict__ feat,
                         const float* __restrict__ rois,
                         float* __restrict__ out) {
    __shared__ float tile[2][TILE_ELEMS];    // double buffer: 92.4 KB

    const int r   = blockIdx.x;
    const int tid = threadIdx.x;

    // ROI decode (block-uniform -> SALU: s_floor_f32 / s_cvt_i32_f32).
    // Coordinates are in-bounds by construction: floor(607.9*0.0625)=37 < 38.
    const int b  = (int)rois[r * 5 + 0];
    const int x1 = (int)floorf(rois[r * 5 + 1] * SCALE);
    const int y1 = (int)floorf(rois[r * 5 + 2] * SCALE);
    const int x2 = (int)floorf(rois[r * 5 + 3] * SCALE);
    const int y2 = (int)floorf(rois[r * 5 + 4] * SCALE);
    const int rh = y2 - y1 + 1;     // >= 1
    const int rw = x2 - x1 + 1;     // >= 1
    const int area  = rh * rw;
    const int nelem = CHUNK * area;

    // ---- prologue: stage chunk 0 into buffer 0 ----
    stage_chunk(feat, b, 0, y1, x1, rh, rw, area, nelem, tile[0], tid);
    wait_async_zero();
    __syncthreads();

    int p = 0;
    for (int c0 = 0; c0 < NCH; c0 += CHUNK) {
        // ---- overlap: kick off async stage of the NEXT chunk ----
        if (c0 + CHUNK < NCH)
            stage_chunk(feat, b, c0 + CHUNK, y1, x1, rh, rw, area, nelem,
                        tile[p ^ 1], tid);

        // ---- compute current chunk's 7x7 bin maxes out of LDS ----
        const float* cur = tile[p];
        for (int t = tid; t < CHUNK * OUT_HW * OUT_HW; t += THREADS) {
            const int cc  = t / (OUT_HW * OUT_HW);
            const int bin = t - cc * (OUT_HW * OUT_HW);
            const int i   = bin / OUT_HW;
            const int j   = bin - i * OUT_HW;

            // reference bin math: hstart = y1 + (i*rh)//7 (relative: drop y1)
            const int h0 = (i * rh) / OUT_HW;
            const int h1 = ((i + 1) * rh + OUT_HW - 1) / OUT_HW;
            const int w0 = (j * rw) / OUT_HW;
            const int w1 = ((j + 1) * rw + OUT_HW - 1) / OUT_HW;

            float m = -__builtin_inff();
            const float* base = cur + cc * area;
            for (int y = h0; y < h1; ++y) {
                const float* rowp = base + y * rw;
                for (int x = w0; x < w1; ++x)
                    m = fmaxf(m, rowp[x]);
            }
            out[(((size_t)r * NCH + (c0 + cc)) * OUT_HW + i) * OUT_HW + j] = m;
        }

        // ---- drain next-chunk stage, make it visible, swap buffers ----
        wait_async_zero();
        __syncthreads();
        p ^= 1;
    }
}

extern "C" void kernel_launch(void* const* d_in, const int* in_sizes, int n_in,
                              void* d_out, int out_size, void* d_ws, size_t ws_size,
                              hipStream_t stream) {
    const float* feat = (const float*)d_in[0];   // [2,256,38,38] f32
    const float* rois = (const float*)d_in[1];   // [256,5] f32
    float* out = (float*)d_out;                  // [256,256,7,7] f32
    (void)in_sizes; (void)n_in; (void)out_size; (void)d_ws; (void)ws_size;

    roi_max_pool_kernel<<<dim3(NUM_ROIS), dim3(THREADS), 0, stream>>>(feat, rois, out);
}